// WLConvContinuous_936302871058
// MI455X (gfx1250) — compile-verified
//
#include <hip/hip_runtime.h>

// ---------------------------------------------------------------------------
// WLConvContinuous: out = 0.5 * (x + segment_mean(x[src], dst))
//   x: [N,32] f32, edge_index: [2,E] int (row0 = src, row1 = dst)
// Memory-bound scatter/gather: edge stream staged to LDS via the CDNA5
// Tensor Data Mover (tensor_load_to_lds + s_wait_tensorcnt), accumulation
// via native global_atomic_add_f32 into the L2-resident output buffer.
// ---------------------------------------------------------------------------

#define FEAT 32
#define EDGE_TILE 1024          // edges per TDM tile (per row); 2 rows -> 8KB/buffer
#define SCATTER_BLOCKS 512
#define SCATTER_THREADS 256

typedef unsigned int u32x4 __attribute__((ext_vector_type(4)));
typedef int i32x8 __attribute__((ext_vector_type(8)));
typedef int i32x4 __attribute__((ext_vector_type(4)));

#if defined(__HIP_DEVICE_COMPILE__) && __has_builtin(__builtin_amdgcn_tensor_load_to_lds)
#define USE_TDM 1
#else
#define USE_TDM 0
#endif

// ---------------------------------------------------------------------------
// Kernel 1: zero the accumulator (d_out reused as sums) and the u32 counts.
// ---------------------------------------------------------------------------
__global__ void wl_zero_kernel(float4* __restrict__ out4,
                               unsigned* __restrict__ cnt,
                               int nOut4, int nCnt) {
  int i = blockIdx.x * blockDim.x + threadIdx.x;
  if (i < nOut4) out4[i] = make_float4(0.f, 0.f, 0.f, 0.f);
  if (i < nCnt)  cnt[i] = 0u;
}

#if USE_TDM
// Issue one TDM 2D tile load: rows {src,dst} x EDGE_TILE int32 edges -> LDS.
// D# layout per CDNA5 ISA ch.8 (groups 0/1; groups 2/3 zero => dims <= 2D).
// clang-23 toolchain: 6-arg builtin
//   (u32x4 g0, i32x8 g1, i32x4 g2, i32x4 g3, i32x8 extra, i32 cpol)
__device__ __forceinline__ void tdm_load_edge_tile(const int* __restrict__ eidx,
                                                   int tileStart, int E,
                                                   unsigned lds_off) {
  unsigned long long ga =
      (unsigned long long)(unsigned long long)(const void*)(eidx + tileStart);
  unsigned remain = (unsigned)(E - tileStart);   // tensor_dim0 for OOB zero-fill

  u32x4 g0;
  g0[0] = 1u;                                            // count=1, user mode
  g0[1] = lds_off;                                       // lds_addr (bytes)
  g0[2] = (unsigned)(ga & 0xFFFFFFFFull);                // global_addr[31:0]
  g0[3] = (unsigned)((ga >> 32) & 0x01FFFFFFull)         // global_addr[56:32]
        | (2u << 30);                                    // type = 2 ("image")

  i32x8 g1;
  g1[0] = (int)(2u << 16);                               // data_size=2 (4B elems)
  g1[1] = (int)((remain & 0xFFFFu) << 16);               // tensor_dim0[15:0]
  g1[2] = (int)((remain >> 16) | (2u << 16));            // dim0[31:16] | tensor_dim1=2
  g1[3] = (int)((unsigned)EDGE_TILE << 16);              // tile_dim0 = EDGE_TILE
  g1[4] = 2;                                             // tile_dim1=2, tile_dim2=0
  g1[5] = (int)(unsigned)E;                              // tensor_dim0_stride = E
  g1[6] = 0;                                             // stride hi / dim1_stride
  g1[7] = 0;

  i32x4 gz4 = {0, 0, 0, 0};                              // groups 2/3: unused dims
  i32x8 gz8 = {0, 0, 0, 0, 0, 0, 0, 0};
  __builtin_amdgcn_tensor_load_to_lds(g0, g1, gz4, gz4, gz8, 0);
}
#endif

// ---------------------------------------------------------------------------
// Kernel 2: edge scatter. 8 lanes per edge (one float4 chunk each): one edge
// gathers exactly one 128B row of x, then 4 no-return f32 atomics into sums.
// Edge indices are DMA-staged into LDS (double buffered) by the TDM.
// ---------------------------------------------------------------------------
__global__ __launch_bounds__(SCATTER_THREADS)
void wl_edge_scatter_kernel(const float* __restrict__ x,
                            const int* __restrict__ eidx,
                            float* __restrict__ sums,
                            unsigned* __restrict__ cnt,
                            int E, int perBlock) {
  __shared__ int ebuf[2][2 * EDGE_TILE];   // [buffer][src row | dst row]

  const int tid = threadIdx.x;
  const int start0 = blockIdx.x * perBlock;
  const int end = min(E, start0 + perBlock);
  if (start0 >= end) return;               // uniform across the block
  const int nt = (end - start0 + EDGE_TILE - 1) / EDGE_TILE;

#if USE_TDM
  const unsigned lds0 = (unsigned)(unsigned long long)(void*)&ebuf[0][0];
  const unsigned lds1 = (unsigned)(unsigned long long)(void*)&ebuf[1][0];
  const bool isWave0 = (tid < 32);
  if (isWave0) tdm_load_edge_tile(eidx, start0, E, lds0);   // prologue: tile 0
#endif

  const int group = tid >> 3;              // 32 edge-groups per block
  const int chunk = tid & 7;               // float4 chunk within the 32-f row

  for (int t = 0; t < nt; ++t) {
    const int cur = t & 1;
    const int tileStart = start0 + t * EDGE_TILE;
    const int tcnt = min(EDGE_TILE, end - tileStart);

#if USE_TDM
    if (isWave0) {
      if (t + 1 < nt) {                    // prefetch next tile, then make sure
        tdm_load_edge_tile(eidx, tileStart + EDGE_TILE, E, cur ? lds0 : lds1);
        __builtin_amdgcn_s_wait_tensorcnt(1);   // tile t complete (in-order TDM)
      } else {
        __builtin_amdgcn_s_wait_tensorcnt(0);
      }
    }
#else
    for (int i = tid; i < EDGE_TILE; i += SCATTER_THREADS) {
      int g = tileStart + i;
      ebuf[cur][i]             = (g < E) ? eidx[g]     : 0;
      ebuf[cur][EDGE_TILE + i] = (g < E) ? eidx[E + g] : 0;
    }
#endif
    __syncthreads();                       // LDS tile visible to all 8 waves

    for (int e = group; e < tcnt; e += SCATTER_THREADS / 8) {
      const int s = ebuf[cur][e];                    // ds_load, broadcast x8
      const int d = ebuf[cur][EDGE_TILE + e];
      const float4 v = ((const float4*)(x + (size_t)s * FEAT))[chunk];  // b128
      float* dp = sums + (size_t)d * FEAT + chunk * 4;
      unsafeAtomicAdd(dp + 0, v.x);                  // global_atomic_add_f32
      unsafeAtomicAdd(dp + 1, v.y);
      unsafeAtomicAdd(dp + 2, v.z);
      unsafeAtomicAdd(dp + 3, v.w);
      if (chunk == 0) atomicAdd(&cnt[d], 1u);        // global_atomic_add_u32
    }
    __syncthreads();                       // buffer free for next TDM overwrite
  }
}

// ---------------------------------------------------------------------------
// Kernel 3: out = 0.5*x + sums * (0.5 / max(cnt,1)), fused, in place on d_out.
// One thread per float4.
// ---------------------------------------------------------------------------
__global__ void wl_finalize_kernel(const float* __restrict__ x,
                                   const unsigned* __restrict__ cnt,
                                   float* __restrict__ out, int n4) {
  int i = blockIdx.x * blockDim.x + threadIdx.x;
  if (i >= n4) return;
  const int node = i >> 3;                 // 8 float4 chunks per node
  const float c = (float)cnt[node];
  const float inv = (c > 0.f) ? (0.5f / c) : 0.5f;   // cnt==0 -> sums==0 anyway
  const float4 s  = ((const float4*)out)[i];
  const float4 xv = ((const float4*)x)[i];
  float4 r;
  r.x = fmaf(s.x, inv, 0.5f * xv.x);
  r.y = fmaf(s.y, inv, 0.5f * xv.y);
  r.z = fmaf(s.z, inv, 0.5f * xv.z);
  r.w = fmaf(s.w, inv, 0.5f * xv.w);
  ((float4*)out)[i] = r;
}

// ---------------------------------------------------------------------------
extern "C" void kernel_launch(void* const* d_in, const int* in_sizes, int n_in,
                              void* d_out, int out_size, void* d_ws, size_t ws_size,
                              hipStream_t stream) {
  const float* x    = (const float*)d_in[0];   // [N, 32] f32
  const int*   eidx = (const int*)d_in[1];     // [2, E] int32
  float*       out  = (float*)d_out;           // [N, 32] f32 (also sum scratch)
  unsigned*    cnt  = (unsigned*)d_ws;         // [N] u32 degree counts

  const int NF = in_sizes[0];                  // N * 32
  const int N  = NF / FEAT;
  const int E  = in_sizes[1] / 2;
  const int nOut4 = NF / 4;

  // 1) zero sums + counts
  {
    int zmax = (nOut4 > N) ? nOut4 : N;
    wl_zero_kernel<<<(zmax + 255) / 256, 256, 0, stream>>>(
        (float4*)out, cnt, nOut4, N);
  }
  // 2) TDM-staged edge scatter with f32 atomics
  {
    int perBlock = (E + SCATTER_BLOCKS - 1) / SCATTER_BLOCKS;
    wl_edge_scatter_kernel<<<SCATTER_BLOCKS, SCATTER_THREADS, 0, stream>>>(
        x, eidx, out, cnt, E, perBlock);
  }
  // 3) fused mean + blend
  wl_finalize_kernel<<<(nOut4 + 255) / 256, 256, 0, stream>>>(x, cnt, out, nOut4);
}